// MiniSpinNet_52243982188918
// MI455X (gfx1250) — compile-verified
//
#include <hip/hip_runtime.h>
#include <hip/hip_bf16.h>
#include <math.h>

typedef __attribute__((ext_vector_type(2))) float v2f;
typedef __attribute__((ext_vector_type(8))) float v8f;
typedef __attribute__((ext_vector_type(4))) int   v4i;

#define PI_F 3.14159265358979323846f

// CDNA5 async global->LDS path (ASYNCcnt), guarded so compile never breaks.
// Probe-derived signature: (global int4*, local int4*, imm offset, imm cpol).
#if defined(__AMDGCN__) && \
    __has_builtin(__builtin_amdgcn_global_load_async_to_lds_b128) && \
    __has_builtin(__builtin_amdgcn_s_wait_asynccnt)
#define HAS_ASYNC_LDS 1
#define GPTR(p) ((__attribute__((address_space(1))) v4i*)(p))
#define LPTR(p) ((__attribute__((address_space(3))) v4i*)(p))
#else
#define HAS_ASYNC_LDS 0
#endif

// ---------------------------------------------------------------------------
// Kernel 0: weight pre-transpose. Bt[n][k] so conv B fragments are contiguous
// (k, k+1) pairs -> one aligned global_load_b64 each, L2-resident.
//   conv3d: k = ((dr*3+de)*3+da)*16 + c   (K=432)
//   conv2d: k = (de*3+da)*32 + c          (K=288)
// ---------------------------------------------------------------------------
__global__ __launch_bounds__(256) void k_prep_weights(
    const float* __restrict__ c3_w, const float* __restrict__ c2a_w,
    const float* __restrict__ c2b_w, float* __restrict__ Bt3,
    float* __restrict__ Bt2a, float* __restrict__ Bt2b)
{
    int i = blockIdx.x * 256 + threadIdx.x;
    if (i < 13824) {
        int n = i / 432, k = i % 432, t = k >> 4, c = k & 15;
        int dr = t / 9, de = (t % 9) / 3, da = t % 3;
        Bt3[i] = c3_w[(((n * 16 + c) * 3 + dr) * 3 + de) * 3 + da];
    }
    if (i < 9216) {
        int n = i / 288, k = i % 288, t = k >> 5, c = k & 31;
        int de = t / 3, da = t % 3;
        int src = ((n * 32 + c) * 3 + de) * 3 + da;
        Bt2a[i] = c2a_w[src];
        Bt2b[i] = c2b_w[src];
    }
}

// ---------------------------------------------------------------------------
// Kernel 1: ball_query. One wave32 per keypoint. Ordered (by index) selection
// of first 512 in-radius points via ballot + prefix popcount; fill = keypoint.
// ---------------------------------------------------------------------------
__global__ __launch_bounds__(32) void k_ball_query(
    const float* __restrict__ pts, const float* __restrict__ kpts,
    float* __restrict__ patch, int N, int Kq)
{
    int m = blockIdx.x;
    int lane = threadIdx.x;
    float kx = kpts[m * 3 + 0], ky = kpts[m * 3 + 1], kz = kpts[m * 3 + 2];
    int base = 0;
    for (int start = 0; start < N && base < Kq; start += 32) {
        int i = start + lane;
        bool valid = false;
        float px = 0.f, py = 0.f, pz = 0.f;
        if (i < N) {
            px = pts[i * 3 + 0]; py = pts[i * 3 + 1]; pz = pts[i * 3 + 2];
            float dx = kx - px, dy = ky - py, dz = kz - pz;
            valid = (dx * dx + dy * dy + dz * dz) < 9.0f; // DES_R^2
        }
        unsigned mlo = (unsigned)__ballot(valid);
        int prefix = __popc(mlo & ((1u << lane) - 1u));
        int slot = base + prefix;
        if (valid && slot < Kq) {
            patch[(m * Kq + slot) * 3 + 0] = px;
            patch[(m * Kq + slot) * 3 + 1] = py;
            patch[(m * Kq + slot) * 3 + 2] = pz;
        }
        base += __popc(mlo);
    }
    int startfill = base < Kq ? base : Kq;
    for (int s = startfill + lane; s < Kq; s += 32) {
        patch[(m * Kq + s) * 3 + 0] = kx;
        patch[(m * Kq + s) * 3 + 1] = ky;
        patch[(m * Kq + s) * 3 + 2] = kz;
    }
}

// ---------------------------------------------------------------------------
// Kernel 2: sphere_query + var_to_invar + point MLP (3->16) + max over 10
// samples. One block per keypoint; patches staged in LDS. Output layout
// x[m][c=16][r=3][e=7][a=20] (matches reference reshape of the 420 voxels).
// ---------------------------------------------------------------------------
__global__ __launch_bounds__(256) void k_group_mlp(
    const float* __restrict__ patch,
    const float* __restrict__ pnt_w, const float* __restrict__ pnt_b,
    const float* __restrict__ pnt_g, const float* __restrict__ pnt_bt,
    float* __restrict__ xout)
{
    __shared__ float sp[512 * 3];
    __shared__ float sw[48], sb[16], sg[16], sbt[16];
    int m = blockIdx.x, tid = threadIdx.x;
    const float* pm = patch + m * 512 * 3;
    float cx = pm[511 * 3 + 0], cy = pm[511 * 3 + 1], cz = pm[511 * 3 + 2];
    const float invR = 1.0f / 3.0f;
    for (int i = tid; i < 512; i += 256) {
        sp[i * 3 + 0] = (pm[i * 3 + 0] - cx) * invR;
        sp[i * 3 + 1] = (pm[i * 3 + 1] - cy) * invR;
        sp[i * 3 + 2] = (pm[i * 3 + 2] - cz) * invR;
    }
    if (tid < 48) sw[tid] = pnt_w[tid];
    if (tid < 16) { sb[tid] = pnt_b[tid]; sg[tid] = pnt_g[tid]; sbt[tid] = pnt_bt[tid]; }
    __syncthreads();

    const float r2 = (0.8f / 3.0f) * (0.8f / 3.0f); // (DELTA/RAD_N)^2
    for (int v = tid; v < 420; v += 256) {
        int ri = v / 140, rem = v % 140, ei = rem / 20, ai = rem % 20;
        float r = (ri + 0.5f) * (1.0f / 3.0f);
        float beta = (ei + 0.5f) * (PI_F / 7.0f);
        float alpha = ai * (2.0f * PI_F / 20.0f);
        float sbe = sinf(beta), cbe = cosf(beta);
        float vx = r * sbe * cosf(alpha);
        float vy = r * sbe * sinf(alpha);
        float vz = r * cbe;
        float ang = -alpha;
        float ca = cosf(ang), sa = sinf(ang);

        float xmax[16];
        #pragma unroll
        for (int o = 0; o < 16; ++o) xmax[o] = -1e30f;

        int cnt = 0;
        for (int i = 0; i < 512 && cnt < 10; ++i) {
            float dx = sp[i * 3 + 0] - vx;
            float dy = sp[i * 3 + 1] - vy;
            float dz = sp[i * 3 + 2] - vz;
            if (dx * dx + dy * dy + dz * dz < r2) {
                ++cnt;
                float gx = ca * dx - sa * dy;
                float gy = sa * dx + ca * dy;
                float gz = dz;
                #pragma unroll
                for (int o = 0; o < 16; ++o) {
                    float acc = sw[o * 3 + 0] * gx + sw[o * 3 + 1] * gy
                              + sw[o * 3 + 2] * gz + sb[o];
                    float val = acc * sg[o] + sbt[o];
                    val = val > 0.f ? val : 0.f;
                    xmax[o] = fmaxf(xmax[o], val);
                }
            }
        }
        if (cnt < 10) { // zero-filled samples still go through bias+bn+relu
            #pragma unroll
            for (int o = 0; o < 16; ++o) {
                float val = sb[o] * sg[o] + sbt[o];
                val = val > 0.f ? val : 0.f;
                xmax[o] = fmaxf(xmax[o], val);
            }
        }
        #pragma unroll
        for (int o = 0; o < 16; ++o)
            xout[(m * 16 + o) * 420 + v] = xmax[o];
    }
}

// ---------------------------------------------------------------------------
// Kernel 3: conv3d (16->32ch, 3x3x3, rad collapses, ele zero-pad, azi cyclic)
// WMMA f32 GEMM, M=140 (9 tiles) x N=32 (2 tiles) x K=432.
// LDS image xp[c=16][dr=3][ez=11][a=20]: rows ez=1..7 hold e=0..6; rows
// 0,8,9,10 are zero -> ele zero-pad AND zero-sink for M-padding lanes
// (e_eff=8). All A loads are unconditional ds_load with immediate offsets.
// ---------------------------------------------------------------------------
__global__ __launch_bounds__(256) void k_conv3d(
    const float* __restrict__ xin_g, const float* __restrict__ Bt, // [32][432]
    const float* __restrict__ b, const float* __restrict__ g,
    const float* __restrict__ bt, float* __restrict__ out)
{
    __shared__ float xp[16 * 3 * 11 * 20]; // 10560 floats = 42.2 KB
    int m = blockIdx.x, tid = threadIdx.x;
    const float* xm = xin_g + m * 6720;

    // zero rows: 48 planes x rows {0,8,9,10} x 20
    for (int i = tid; i < 48 * 4 * 20; i += 256) {
        int plane = i / 80, r4 = (i / 20) % 4, az = i % 20;
        int ez = (r4 == 0) ? 0 : (7 + r4);
        xp[(plane * 11 + ez) * 20 + az] = 0.f;
    }
    // interior: 48 planes x 140 contiguous floats each (src & dst contiguous)
#if HAS_ASYNC_LDS
    for (int j = tid * 4; j < 6720; j += 1024) {
        int plane = j / 140, rem = j % 140;
        __builtin_amdgcn_global_load_async_to_lds_b128(
            GPTR(xm + j), LPTR(&xp[(plane * 11 + 1) * 20 + rem]), 0, 0);
    }
    __builtin_amdgcn_s_wait_asynccnt(0);
#else
    for (int j = tid; j < 6720; j += 256) {
        int plane = j / 140, rem = j % 140;
        xp[(plane * 11 + 1) * 20 + rem] = xm[j];
    }
#endif
    __builtin_prefetch(Bt, 0, 3);
    __syncthreads();

    int wave = tid >> 5, lane = tid & 31;
    int lo = lane & 15, hi = lane >> 4;
    int hioff = hi * 2 * 660; // this lane's K-pair channel offset (c stride 660)

    for (int tile = wave; tile < 18; tile += 8) {
        int n0 = (tile & 1) * 16;
        int p0 = (tile >> 1) * 16;
        int p = p0 + lo;
        bool pv = p < 140;
        int e_eff = pv ? (p / 20) : 8; // e_eff=8 -> rows 8,9,10 (all zero)
        int a_eff = pv ? (p % 20) : 0;
        int n = n0 + lo;
        const float* bp = Bt + n * 432 + 2 * hi;
        v8f acc = {0.f, 0.f, 0.f, 0.f, 0.f, 0.f, 0.f, 0.f};

        int kbase = 0;
        for (int dr = 0; dr < 3; ++dr) {
            for (int de = 0; de < 3; ++de) {
                int rowbase = dr * 220 + (e_eff + de) * 20 + hioff;
                for (int da = 0; da < 3; ++da) {
                    int aa = a_eff + da - 1;
                    aa = aa < 0 ? 19 : (aa > 19 ? 0 : aa); // cyclic azi
                    int abase = rowbase + aa;
                    #pragma unroll
                    for (int ch = 0; ch < 4; ++ch) {
                        v2f av;
                        av.x = xp[abase + ch * 2640];       // c = ch*4+2hi
                        av.y = xp[abase + ch * 2640 + 660]; // c+1
                        v2f bv = *(const v2f*)(bp + kbase + ch * 4);
                        acc = __builtin_amdgcn_wmma_f32_16x16x4_f32(
                            false, av, false, bv, (short)0, acc, false, false);
                    }
                    kbase += 16;
                }
            }
        }
        float bn_b = b[n], bn_g = g[n], bn_bt = bt[n];
        #pragma unroll
        for (int j = 0; j < 8; ++j) {
            int row = p0 + j + 8 * hi; // C/D layout
            if (row < 140) {
                float val = (acc[j] + bn_b) * bn_g + bn_bt;
                val = val > 0.f ? val : 0.f;
                out[(m * 32 + n) * 140 + row] = val;
            }
        }
    }
}

// ---------------------------------------------------------------------------
// Kernel 4: conv2d_cyl (32->32, 3x3, ele zero-pad, azi cyclic) + bn_relu,
// WMMA f32 GEMM: M=140, N=32, K=288. Same padded-image scheme (c stride 220).
// ---------------------------------------------------------------------------
__global__ __launch_bounds__(256) void k_conv2d(
    const float* __restrict__ in_g, const float* __restrict__ Bt, // [32][288]
    const float* __restrict__ b, const float* __restrict__ g,
    const float* __restrict__ bt, float* __restrict__ out)
{
    __shared__ float xp[32 * 11 * 20]; // 7040 floats = 28.2 KB
    int m = blockIdx.x, tid = threadIdx.x;
    const float* im = in_g + m * 4480;

    for (int i = tid; i < 32 * 4 * 20; i += 256) {
        int c = i / 80, r4 = (i / 20) % 4, az = i % 20;
        int ez = (r4 == 0) ? 0 : (7 + r4);
        xp[(c * 11 + ez) * 20 + az] = 0.f;
    }
#if HAS_ASYNC_LDS
    for (int j = tid * 4; j < 4480; j += 1024) {
        int c = j / 140, rem = j % 140;
        __builtin_amdgcn_global_load_async_to_lds_b128(
            GPTR(im + j), LPTR(&xp[(c * 11 + 1) * 20 + rem]), 0, 0);
    }
    __builtin_amdgcn_s_wait_asynccnt(0);
#else
    for (int j = tid; j < 4480; j += 256) {
        int c = j / 140, rem = j % 140;
        xp[(c * 11 + 1) * 20 + rem] = im[j];
    }
#endif
    __builtin_prefetch(Bt, 0, 3);
    __syncthreads();

    int wave = tid >> 5, lane = tid & 31;
    int lo = lane & 15, hi = lane >> 4;
    int hioff = hi * 2 * 220;

    for (int tile = wave; tile < 18; tile += 8) {
        int n0 = (tile & 1) * 16;
        int p0 = (tile >> 1) * 16;
        int p = p0 + lo;
        bool pv = p < 140;
        int e_eff = pv ? (p / 20) : 8;
        int a_eff = pv ? (p % 20) : 0;
        int n = n0 + lo;
        const float* bp = Bt + n * 288 + 2 * hi;
        v8f acc = {0.f, 0.f, 0.f, 0.f, 0.f, 0.f, 0.f, 0.f};

        int kbase = 0;
        for (int de = 0; de < 3; ++de) {
            int rowbase = (e_eff + de) * 20 + hioff;
            for (int da = 0; da < 3; ++da) {
                int aa = a_eff + da - 1;
                aa = aa < 0 ? 19 : (aa > 19 ? 0 : aa);
                int abase = rowbase + aa;
                #pragma unroll
                for (int ch = 0; ch < 8; ++ch) {
                    v2f av;
                    av.x = xp[abase + ch * 880];       // c = ch*4+2hi
                    av.y = xp[abase + ch * 880 + 220]; // c+1
                    v2f bv = *(const v2f*)(bp + kbase + ch * 4);
                    acc = __builtin_amdgcn_wmma_f32_16x16x4_f32(
                        false, av, false, bv, (short)0, acc, false, false);
                }
                kbase += 32;
            }
        }
        float bn_b = b[n], bn_g = g[n], bn_bt = bt[n];
        #pragma unroll
        for (int j = 0; j < 8; ++j) {
            int row = p0 + j + 8 * hi;
            if (row < 140) {
                float val = (acc[j] + bn_b) * bn_g + bn_bt;
                val = val > 0.f ? val : 0.f;
                out[(m * 32 + n) * 140 + row] = val;
            }
        }
    }
}

// ---------------------------------------------------------------------------
// Kernel 5: attention head + normalization. One block per keypoint.
// desc = normalize(mean(x2 * w2)), equi = x2 / ||x2||_channel.
// Deterministic sequential reductions (no float atomics).
// ---------------------------------------------------------------------------
__global__ __launch_bounds__(256) void k_head(
    const float* __restrict__ x2g,
    const float* __restrict__ p1w, const float* __restrict__ p1b,
    const float* __restrict__ p1g, const float* __restrict__ p1bt,
    const float* __restrict__ p2w, const float* __restrict__ p2b,
    const float* __restrict__ p2g, const float* __restrict__ p2bt,
    float* __restrict__ out, int Mtot)
{
    __shared__ float sx[32 * 140];
    __shared__ float ss[140];
    __shared__ float sf[32];
    __shared__ float snorm;
    int m = blockIdx.x, tid = threadIdx.x;
    const float* xm = x2g + m * 4480;
#if HAS_ASYNC_LDS
    for (int j = tid * 4; j < 4480; j += 1024)
        __builtin_amdgcn_global_load_async_to_lds_b128(
            GPTR(xm + j), LPTR(&sx[j]), 0, 0);
    __builtin_amdgcn_s_wait_asynccnt(0);
#else
    for (int i = tid; i < 4480; i += 256) sx[i] = xm[i];
#endif
    __syncthreads();

    if (tid < 140) {
        int p = tid;
        float t1[16];
        #pragma unroll
        for (int o = 0; o < 16; ++o) {
            float acc = p1b[o];
            for (int c = 0; c < 32; ++c) acc += p1w[o * 32 + c] * sx[c * 140 + p];
            acc = acc * p1g[o] + p1bt[o];
            t1[o] = acc > 0.f ? acc : 0.f;
        }
        float acc2 = p2b[0];
        #pragma unroll
        for (int o = 0; o < 16; ++o) acc2 += p2w[o] * t1[o];
        acc2 = acc2 * p2g[0] + p2bt[0];
        ss[p] = acc2 > 0.f ? acc2 : 0.f;
    }
    __syncthreads();

    if (tid < 32) {
        float fs = 0.f;
        for (int p = 0; p < 140; ++p) fs += sx[tid * 140 + p] * ss[p];
        sf[tid] = fs * (1.0f / 140.0f);
    }
    __syncthreads();
    if (tid == 0) {
        float nn = 0.f;
        for (int c = 0; c < 32; ++c) nn += sf[c] * sf[c];
        snorm = fmaxf(sqrtf(nn), 1e-12f);
    }
    __syncthreads();
    if (tid < 32) out[m * 32 + tid] = sf[tid] / snorm;

    const int equi_off = Mtot * 32;
    for (int p = tid; p < 140; p += 256) {
        float nn = 0.f;
        for (int c = 0; c < 32; ++c) { float v = sx[c * 140 + p]; nn += v * v; }
        float inv = 1.0f / fmaxf(sqrtf(nn), 1e-12f);
        for (int c = 0; c < 32; ++c)
            out[equi_off + (m * 32 + c) * 140 + p] = sx[c * 140 + p] * inv;
    }
}

// ---------------------------------------------------------------------------
extern "C" void kernel_launch(void* const* d_in, const int* in_sizes, int n_in,
                              void* d_out, int out_size, void* d_ws, size_t ws_size,
                              hipStream_t stream) {
    const float* pts   = (const float*)d_in[0];   // (1, 20000, 3)
    const float* kpts  = (const float*)d_in[1];   // (1, 512, 3)
    const float* pnt_w = (const float*)d_in[2];
    const float* pnt_b = (const float*)d_in[3];
    const float* pnt_g = (const float*)d_in[4];
    const float* pnt_bt= (const float*)d_in[5];
    const float* c3_w  = (const float*)d_in[6];
    const float* c3_b  = (const float*)d_in[7];
    const float* c3_g  = (const float*)d_in[8];
    const float* c3_bt = (const float*)d_in[9];
    const float* c2a_w = (const float*)d_in[10];
    const float* c2a_b = (const float*)d_in[11];
    const float* c2a_g = (const float*)d_in[12];
    const float* c2a_bt= (const float*)d_in[13];
    const float* c2b_w = (const float*)d_in[14];
    const float* c2b_b = (const float*)d_in[15];
    const float* c2b_g = (const float*)d_in[16];
    const float* c2b_bt= (const float*)d_in[17];
    const float* p1_w  = (const float*)d_in[18];
    const float* p1_b  = (const float*)d_in[19];
    const float* p1_g  = (const float*)d_in[20];
    const float* p1_bt = (const float*)d_in[21];
    const float* p2_w  = (const float*)d_in[22];
    const float* p2_b  = (const float*)d_in[23];
    const float* p2_g  = (const float*)d_in[24];
    const float* p2_bt = (const float*)d_in[25];

    const int M = 512, N = 20000, Kq = 512;
    float* ws = (float*)d_ws;
    float* patch = ws;                         // 512*512*3   =   786432 f
    float* x     = patch + 786432;             // 512*16*420  =  3440640 f
    float* mid   = x + 3440640;                // 512*32*140  =  2293760 f
    float* h2    = mid + 2293760;              // 512*32*140  =  2293760 f
    float* x2    = h2 + 2293760;               // 512*32*140  =  2293760 f
    float* Bt3   = x2 + 2293760;               // 32*432      =    13824 f
    float* Bt2a  = Bt3 + 13824;                // 32*288      =     9216 f
    float* Bt2b  = Bt2a + 9216;                // 32*288      =     9216 f
    // total ~44.6 MB of workspace

    k_prep_weights<<<54, 256, 0, stream>>>(c3_w, c2a_w, c2b_w, Bt3, Bt2a, Bt2b);
    k_ball_query<<<M, 32, 0, stream>>>(pts, kpts, patch, N, Kq);
    k_group_mlp<<<M, 256, 0, stream>>>(patch, pnt_w, pnt_b, pnt_g, pnt_bt, x);
    k_conv3d<<<M, 256, 0, stream>>>(x, Bt3, c3_b, c3_g, c3_bt, mid);
    k_conv2d<<<M, 256, 0, stream>>>(mid, Bt2a, c2a_b, c2a_g, c2a_bt, h2);
    k_conv2d<<<M, 256, 0, stream>>>(h2, Bt2b, c2b_b, c2b_g, c2b_bt, x2);
    k_head<<<M, 256, 0, stream>>>(x2, p1_w, p1_b, p1_g, p1_bt,
                                  p2_w, p2_b, p2_g, p2_bt,
                                  (float*)d_out, M);
}